// GWT_SNN_87273735455055
// MI455X (gfx1250) — compile-verified
//
#include <hip/hip_runtime.h>
#include <hip/hip_bf16.h>

typedef __attribute__((ext_vector_type(2))) float v2f;
typedef __attribute__((ext_vector_type(8))) float v8f;

#define B_TOT   16384
#define IN_F    784
#define HID     128
#define NOUT    512      // NUM_MODULES * HIDDEN
#define TSTEPS  20
#define DECAY_F 0.25f
#define WS_STRIDE 132    // 132 mod 64 == 4 -> conflict-free LDS column reads

// scratch offsets in floats
#define OFF_XT    0u
#define OFF_WXT   12845056u   // 784*16384
#define OFF_WWST  13246464u   // + 784*512
#define OFF_PREX  13312000u   // + 128*512

__device__ __forceinline__ v8f wmma_f32(v2f a, v2f b, v8f c) {
    // D = A(16x4, f32) * B(4x16, f32) + C(16x16, f32)
    return __builtin_amdgcn_wmma_f32_16x16x4_f32(
        false, a, false, b, (short)0, c, false, false);
}

// ---------------- Kernel 0a: x[B][784] -> xT[784][B] (LDS-tiled) -------------
__global__ void k_transpose_x(const float* __restrict__ x, float* __restrict__ xT) {
    __shared__ float t[16][17];
    int tx = threadIdx.x, ty = threadIdx.y;
    int b = blockIdx.y * 16 + ty;
    int k = blockIdx.x * 16 + tx;
    t[ty][tx] = x[b * IN_F + k];
    __syncthreads();
    int ko = blockIdx.x * 16 + ty;
    int bo = blockIdx.y * 16 + tx;
    xT[ko * B_TOT + bo] = t[tx][ty];
}

// ------------- Kernel 0b: W[512][912] -> WxT[784][512], WwsT[128][512] -------
__global__ void k_pack_w(const float* __restrict__ W,
                         float* __restrict__ WxT, float* __restrict__ WwsT) {
    int idx = blockIdx.x * 256 + threadIdx.x;
    if (idx >= NOUT * (IN_F + HID)) return;
    int n = idx / (IN_F + HID);
    int k = idx - n * (IN_F + HID);
    float v = W[idx];
    if (k < IN_F) WxT[k * NOUT + n] = v;
    else          WwsT[(k - IN_F) * NOUT + n] = v;
}

// ------- Kernel 1: pre_x[B][512] = x @ Wx^T + bias  (fp32 WMMA GEMM) ---------
// Each wave computes a 16x64 output strip (4 C tiles), K=784 in chunks of 4.
__global__ void k_prex(const float* __restrict__ xT, const float* __restrict__ WxT,
                       const float* __restrict__ bias, float* __restrict__ preX) {
    int tid  = threadIdx.x;
    int lane = tid & 31, l15 = lane & 15, half = lane >> 4;
    int gw = blockIdx.x * 8 + (tid >> 5);
    int b0 = (gw >> 3) * 16;
    int n0 = (gw & 7) * 64;

    v8f acc[4] = {{}, {}, {}, {}};
    for (int k = 0; k < IN_F; k += 4) {
        const float* xp = xT + (size_t)(k + 2 * half) * B_TOT + b0 + l15;
        v2f A = { xp[0], xp[B_TOT] };                 // K = 2*half, 2*half+1
        const float* wp = WxT + (k + 2 * half) * NOUT + n0 + l15;
#pragma unroll
        for (int t = 0; t < 4; ++t) {
            v2f Bf = { wp[t * 16], wp[NOUT + t * 16] };
            acc[t] = wmma_f32(A, Bf, acc[t]);
        }
    }
#pragma unroll
    for (int t = 0; t < 4; ++t) {
        int col = n0 + t * 16 + l15;
        float bb = bias[col];
#pragma unroll
        for (int i = 0; i < 8; ++i) {
            int row = b0 + i + 8 * half;              // C/D layout: M = i + 8*half
            preX[row * NOUT + col] = acc[t][i] + bb;
        }
    }
}

// --------- Kernel 2: fused 20-step SNN loop, one block per 16 batch rows -----
__global__ void k_snn(const float* __restrict__ preX, const float* __restrict__ WwsT,
                      const float* __restrict__ Wa, const float* __restrict__ Wo,
                      const float* __restrict__ bo, float* __restrict__ out,
                      float* __restrict__ wsum_out) {
    __shared__ float wsb[2][16 * WS_STRIDE];   // ping-pong workspace [16 rows][128 h]
    __shared__ float scores[64];               // [module][b]
    __shared__ float wts[64];                  // softmax weights [module][b]

    int tid  = threadIdx.x;
    int lane = tid & 31, l15 = lane & 15, half = lane >> 4;
    int w    = tid >> 5;                       // wave 0..7 -> cols n0 = w*64
    int b0   = blockIdx.x * 16;
    int n0   = w * 64;
    int mod  = w >> 1;                         // module id for this wave

    // Persistent registers: pre_x tile + Wa fragment + membrane + trace
    v8f prex[4];
    float wa[4];
#pragma unroll
    for (int t = 0; t < 4; ++t) {
        int col = n0 + t * 16 + l15;
        wa[t] = Wa[col & (HID - 1)];
#pragma unroll
        for (int i = 0; i < 8; ++i)
            prex[t][i] = preX[(b0 + i + 8 * half) * NOUT + col];
    }
    v8f mem[4] = {{}, {}, {}, {}};
    float tr[8] = {0.f, 0.f, 0.f, 0.f, 0.f, 0.f, 0.f, 0.f};

    for (int j = tid; j < 16 * WS_STRIDE; j += 256) wsb[0][j] = 0.f;
    __syncthreads();

    int p = 0;
    for (int step = 0; step < TSTEPS; ++step) {
        float* wsC = wsb[p];
        float* wsN = wsb[p ^ 1];

        // Phase A: zero next workspace + scores; recurrent GEMM; LIF update
        for (int j = tid; j < 16 * WS_STRIDE; j += 256) wsN[j] = 0.f;
        if (tid < 64) scores[tid] = 0.f;

        v8f acc[4];
#pragma unroll
        for (int t = 0; t < 4; ++t) acc[t] = prex[t];
        for (int k = 0; k < HID; k += 4) {
            v2f A = { wsC[l15 * WS_STRIDE + k + 2 * half],
                      wsC[l15 * WS_STRIDE + k + 1 + 2 * half] };
            const float* wp = WwsT + (k + 2 * half) * NOUT + n0 + l15;
#pragma unroll
            for (int t = 0; t < 4; ++t) {
                v2f Bf = { wp[t * 16], wp[NOUT + t * 16] };
                acc[t] = wmma_f32(A, Bf, acc[t]);
            }
        }
        // mem = mem*decay + pre; spike; reset. Reuse acc[] as spike storage.
#pragma unroll
        for (int t = 0; t < 4; ++t)
#pragma unroll
            for (int i = 0; i < 8; ++i) {
                float m = mem[t][i] * DECAY_F + acc[t][i];
                float s = (m - 1.0f > 0.0f) ? 1.0f : 0.0f;
                mem[t][i] = m * (1.0f - s);
                acc[t][i] = s;                 // spike
            }
        __syncthreads();

        // Phase B: scores[m][b] += spk * Wa[h]  (spike is 0/1 -> add wa only)
#pragma unroll
        for (int t = 0; t < 4; ++t)
#pragma unroll
            for (int i = 0; i < 8; ++i)
                if (acc[t][i] != 0.0f)
                    atomicAdd(&scores[mod * 16 + i + 8 * half], wa[t]);
        __syncthreads();

        // Phase C: softmax over 4 modules per batch row (ba cancels)
        if (tid < 16) {
            float s0 = scores[tid],      s1 = scores[16 + tid];
            float s2 = scores[32 + tid], s3 = scores[48 + tid];
            float mx = fmaxf(fmaxf(s0, s1), fmaxf(s2, s3));
            float e0 = __expf(s0 - mx), e1 = __expf(s1 - mx);
            float e2 = __expf(s2 - mx), e3 = __expf(s3 - mx);
            float inv = 1.0f / (e0 + e1 + e2 + e3);
            wts[tid] = e0 * inv; wts[16 + tid] = e1 * inv;
            wts[32 + tid] = e2 * inv; wts[48 + tid] = e3 * inv;
        }
        __syncthreads();

        // Phase D: ws_new[b][h] = sum_m wts[m][b]*spk[m][b][h]
#pragma unroll
        for (int t = 0; t < 4; ++t) {
            int hl = (n0 + t * 16 + l15) & (HID - 1);
#pragma unroll
            for (int i = 0; i < 8; ++i)
                if (acc[t][i] != 0.0f) {
                    int b = i + 8 * half;
                    atomicAdd(&wsN[b * WS_STRIDE + hl], wts[mod * 16 + b]);
                }
        }
        __syncthreads();

        // Phase E: accumulate time trace in registers (8 elems per thread)
#pragma unroll
        for (int j = 0; j < 8; ++j) {
            int e = tid * 8 + j;
            tr[j] += wsN[(e >> 7) * WS_STRIDE + (e & 127)];
        }
        p ^= 1;
    }
    __syncthreads();

    // workspace_sum = trace mean; write output #2 and stage for final GEMM
    const float inv_t = 1.0f / (float)TSTEPS;
#pragma unroll
    for (int j = 0; j < 8; ++j) {
        int e = tid * 8 + j;
        int b = e >> 7, h = e & 127;
        float v = tr[j] * inv_t;
        wsum_out[(b0 + b) * HID + h] = v;
        wsb[0][b * WS_STRIDE + h] = v;
    }
    __syncthreads();

    // out[b][o] = wsum[b] . Wo[o] + bo[o]   (16x10 tiny projection)
    if (tid < 160) {
        int b = tid / 10, o = tid - b * 10;
        float s = bo[o];
        for (int h = 0; h < HID; ++h)
            s += wsb[0][b * WS_STRIDE + h] * Wo[o * HID + h];
        out[(b0 + b) * 10 + o] = s;
    }
}

extern "C" void kernel_launch(void* const* d_in, const int* in_sizes, int n_in,
                              void* d_out, int out_size, void* d_ws, size_t ws_size,
                              hipStream_t stream) {
    const float* x  = (const float*)d_in[0];   // [16384,784]
    const float* W  = (const float*)d_in[1];   // [4,128,912]
    const float* b  = (const float*)d_in[2];   // [4,128] == bias[512]
    const float* Wa = (const float*)d_in[3];   // [128]
    // d_in[4] = ba : cancels under softmax, unused
    const float* Wo = (const float*)d_in[5];   // [10,128]
    const float* bo = (const float*)d_in[6];   // [10]

    float* ws   = (float*)d_ws;
    float* xT   = ws + OFF_XT;
    float* WxT  = ws + OFF_WXT;
    float* WwsT = ws + OFF_WWST;
    float* preX = ws + OFF_PREX;

    float* out_logits = (float*)d_out;                 // [16384,10]
    float* out_wsum   = (float*)d_out + B_TOT * 10;    // [16384,128]

    dim3 tb(16, 16);
    dim3 tg(IN_F / 16, B_TOT / 16);                    // 49 x 1024
    k_transpose_x<<<tg, tb, 0, stream>>>(x, xT);

    int wtot = NOUT * (IN_F + HID);
    k_pack_w<<<(wtot + 255) / 256, 256, 0, stream>>>(W, WxT, WwsT);

    // 8192 waves = (1024 row strips) x (8 col groups); 8 waves per block
    k_prex<<<1024, 256, 0, stream>>>(xT, WxT, b, preX);

    // one block per 16 batch rows, fused 20-step recurrence
    k_snn<<<B_TOT / 16, 256, 0, stream>>>(preX, WwsT, Wa, Wo, bo,
                                          out_logits, out_wsum);
}